// TransformerMessagePasser_15899968930395
// MI455X (gfx1250) — compile-verified
//
#include <hip/hip_runtime.h>
#include <math.h>

typedef __attribute__((ext_vector_type(2))) float v2f;
typedef __attribute__((ext_vector_type(8))) float v8f;

#define DMODEL 256
#define NHEADS 8

// ---------- ordered-float helpers for atomic max on unsigned ----------
__device__ __forceinline__ unsigned f2ord(float f) {
  unsigned u = __float_as_uint(f);
  return (u & 0x80000000u) ? ~u : (u | 0x80000000u);
}
__device__ __forceinline__ float ord2f(unsigned u) {
  return (u & 0x80000000u) ? __uint_as_float(u & 0x7fffffffu)
                           : __uint_as_float(~u);
}

// ---------- fuse W[256,256] with per-head 32x32 transform T[8,32,32] ----------
// FW[d, h*32+j] = sum_i W[d, h*32+i] * T[h,i,j]; FB likewise from bias.
__global__ __launch_bounds__(256) void fuse_weights(
    const float* __restrict__ W, const float* __restrict__ bias,
    const float* __restrict__ T, float* __restrict__ FW, float* __restrict__ FB) {
  int j = blockIdx.x * blockDim.x + threadIdx.x;
  if (j >= DMODEL * DMODEL) return;
  int d = j >> 8;
  int c = j & 255;
  int h = c >> 5;
  int jj = c & 31;
  const float* wrow = W + (size_t)d * DMODEL + h * 32;
  const float* tcol = T + h * 1024 + jj;  // T[h,i,jj], stride 32 over i
  float acc = 0.f;
#pragma unroll
  for (int i = 0; i < 32; ++i) acc = fmaf(wrow[i], tcol[i * 32], acc);
  FW[j] = acc;
  if (d == 0) {
    const float* brow = bias + h * 32;
    float bacc = 0.f;
#pragma unroll
    for (int i = 0; i < 32; ++i) bacc = fmaf(brow[i], tcol[i * 32], bacc);
    FB[c] = bacc;
  }
}

// ---------- fp32 WMMA GEMM: C[M,256] = X[M,256] @ W[256,256] + bias ----------
// 4 waves / block; each wave owns one 16-row M tile and a 64-wide N group.
__global__ __launch_bounds__(128) void gemm_wmma_f32(
    const float* __restrict__ X, const float* __restrict__ W,
    const float* __restrict__ bias, float* __restrict__ C, int M) {
  const int lane = threadIdx.x & 31;
  const int wid = threadIdx.x >> 5;
  const int mtile = blockIdx.x * 4 + wid;
  const int mtiles = (M + 15) >> 4;
  if (mtile >= mtiles) return;  // whole-wave exit: EXEC stays all-1s for WMMA
  const int m0 = mtile * 16;
  const int ng = blockIdx.y;  // 0..3 -> 64-wide column group
  const int l15 = lane & 15;
  const int hi = lane >> 4;        // 0 for lanes 0-15, 1 for 16-31
  const int koff = hi * 2;         // A/B fragments: lanes>=16 carry K+2,K+3
  int row = m0 + l15;
  int rowc = row < M ? row : (M - 1);
  const float* xrow = X + (size_t)rowc * DMODEL;
  const int nbase = ng * 64;

  v8f acc[4] = {};
#pragma unroll 4
  for (int k0 = 0; k0 < DMODEL; k0 += 4) {
    v2f a;
    a.x = xrow[k0 + koff];
    a.y = xrow[k0 + koff + 1];
#pragma unroll
    for (int s = 0; s < 4; ++s) {
      int col = nbase + s * 16 + l15;
      v2f b;
      b.x = W[(size_t)(k0 + koff) * DMODEL + col];
      b.y = W[(size_t)(k0 + koff + 1) * DMODEL + col];
      acc[s] = __builtin_amdgcn_wmma_f32_16x16x4_f32(
          false, a, false, b, (short)0, acc[s], false, false);
    }
  }
  // D layout: lanes 0-15 -> rows m0+r (VGPR r), lanes 16-31 -> rows m0+8+r
  const int rbase = hi * 8;
#pragma unroll
  for (int s = 0; s < 4; ++s) {
    int col = nbase + s * 16 + l15;
    float bv = bias[col];
#pragma unroll
    for (int r = 0; r < 8; ++r) {
      int mrow = m0 + rbase + r;
      if (mrow < M) C[(size_t)mrow * DMODEL + col] = acc[s][r] + bv;
    }
  }
}

// ---------- zero init for softmax tables ----------
__global__ __launch_bounds__(256) void zero_u32(unsigned* __restrict__ p, int n) {
  int i = blockIdx.x * blockDim.x + threadIdx.x;
  if (i < n) p[i] = 0u;
}

// ---------- per-edge attention scores + segment max ----------
// one wave per edge; 4 lanes per head, float4 loads, shfl reduce.
__global__ __launch_bounds__(256) void edge_scores(
    const float* __restrict__ q, const float* __restrict__ k,
    const int* __restrict__ src, const int* __restrict__ dst,
    const float* __restrict__ pri, float scale, int E,
    float* __restrict__ t_out, unsigned* __restrict__ m_ord) {
  int gid = blockIdx.x * blockDim.x + threadIdx.x;
  int e = gid >> 5;
  int lane = gid & 31;
  if (e >= E) return;
  int s = src[e], d = dst[e];
  int h = lane >> 2, c = lane & 3;
  const float4* qv = (const float4*)(q + (size_t)d * DMODEL + h * 32) + c * 2;
  const float4* kv = (const float4*)(k + (size_t)s * DMODEL + h * 32) + c * 2;
  float4 q0 = qv[0], q1 = qv[1];
  float4 k0 = kv[0], k1 = kv[1];
  float p = q0.x * k0.x + q0.y * k0.y + q0.z * k0.z + q0.w * k0.w +
            q1.x * k1.x + q1.y * k1.y + q1.z * k1.z + q1.w * k1.w;
  p += __shfl_xor(p, 1, 32);
  p += __shfl_xor(p, 2, 32);
  if (c == 0) {
    float t = p * pri[h] * scale;
    t_out[(size_t)e * NHEADS + h] = t;
    atomicMax(&m_ord[(size_t)d * NHEADS + h], f2ord(t));
  }
}

// ---------- exp(t - m) in place + segment sum ----------
__global__ __launch_bounds__(256) void edge_exp(
    const int* __restrict__ dst, const unsigned* __restrict__ m_ord,
    float* __restrict__ t, float* __restrict__ s_sum, int E) {
  int gid = blockIdx.x * blockDim.x + threadIdx.x;
  if (gid >= E * NHEADS) return;
  int e = gid >> 3, h = gid & 7;
  int d = dst[e];
  float m = ord2f(m_ord[(size_t)d * NHEADS + h]);
  float v = __expf(t[gid] - m);
  t[gid] = v;
  atomicAdd(&s_sum[(size_t)d * NHEADS + h], v);
}

// ---------- normalize ----------
__global__ __launch_bounds__(256) void edge_norm(
    const int* __restrict__ dst, const float* __restrict__ s_sum,
    float* __restrict__ t, int E) {
  int gid = blockIdx.x * blockDim.x + threadIdx.x;
  if (gid >= E * NHEADS) return;
  int e = gid >> 3, h = gid & 7;
  t[gid] = t[gid] / s_sum[(size_t)dst[e] * NHEADS + h];
}

extern "C" void kernel_launch(void* const* d_in, const int* in_sizes, int n_in,
                              void* d_out, int out_size, void* d_ws, size_t ws_size,
                              hipStream_t stream) {
  const float* h_video  = (const float*)d_in[0];
  const float* h_tag    = (const float*)d_in[1];
  const float* h0_video = (const float*)d_in[2];
  const float* h0_tag   = (const float*)d_in[3];
  const float* Wk_video = (const float*)d_in[4];
  const float* bk_video = (const float*)d_in[5];
  const float* Wv_video = (const float*)d_in[6];
  const float* bv_video = (const float*)d_in[7];
  const float* Wk_tag   = (const float*)d_in[8];
  const float* bk_tag   = (const float*)d_in[9];
  const float* Wv_tag   = (const float*)d_in[10];
  const float* bv_tag   = (const float*)d_in[11];
  const float* Wq_video = (const float*)d_in[12];
  const float* bq_video = (const float*)d_in[13];
  const float* Wq_tag   = (const float*)d_in[14];
  const float* bq_tag   = (const float*)d_in[15];
  const float* rel_att  = (const float*)d_in[16];  // [3,8,32,32]
  const float* rel_msg  = (const float*)d_in[17];  // [3,8,32,32]
  const float* rel_pri  = (const float*)d_in[18];  // [3,8]
  const int* fb_src = (const int*)d_in[19];
  const int* fb_dst = (const int*)d_in[20];
  const int* st_src = (const int*)d_in[21];
  const int* st_dst = (const int*)d_in[22];
  const int* ht_src = (const int*)d_in[23];
  const int* ht_dst = (const int*)d_in[24];

  const int Nv = in_sizes[0] / DMODEL;
  const int Nt = in_sizes[1] / DMODEL;
  const int E_fb = in_sizes[19], E_st = in_sizes[21], E_ht = in_sizes[23];

  float* out = (float*)d_out;
  const size_t O_tfb = 0;
  const size_t O_tst = O_tfb + (size_t)E_fb * NHEADS;
  const size_t O_tht = O_tst + (size_t)E_st * NHEADS;
  const size_t O_vfb = O_tht + (size_t)E_ht * NHEADS;
  const size_t O_vst = O_vfb + (size_t)Nv * DMODEL;
  const size_t O_vht = O_vst + (size_t)Nt * DMODEL;

  // workspace layout (floats)
  float* ws = (float*)d_ws;
  float* FW   = ws;                                   // 6 * 256*256
  float* FB   = FW + 6 * DMODEL * DMODEL;             // 6 * 256
  float* k_fb = FB + 6 * DMODEL;
  float* q_v  = k_fb + (size_t)Nv * DMODEL;
  float* k_st = q_v  + (size_t)Nv * DMODEL;
  float* q_t  = k_st + (size_t)Nt * DMODEL;
  float* k_ht = q_t  + (size_t)Nt * DMODEL;
  unsigned* m_v = (unsigned*)(k_ht + (size_t)Nv * DMODEL);  // Nv*8 u32
  float*    s_v = (float*)(m_v + (size_t)Nv * NHEADS);      // Nv*8 f32
  unsigned* m_t = (unsigned*)(s_v + (size_t)Nv * NHEADS);   // Nt*8 u32
  float*    s_t = (float*)(m_t + (size_t)Nt * NHEADS);      // Nt*8 f32

  const int TSLICE = NHEADS * 32 * 32;  // one relation slice of rel_att/rel_msg
  // 1) fused weights: 0:fb_k 1:fb_v 2:st_k 3:st_v 4:ht_k 5:ht_v
  fuse_weights<<<256, 256, 0, stream>>>(Wk_video, bk_video, rel_att + 0 * TSLICE, FW + 0 * 65536, FB + 0 * 256);
  fuse_weights<<<256, 256, 0, stream>>>(Wv_video, bv_video, rel_msg + 0 * TSLICE, FW + 1 * 65536, FB + 1 * 256);
  fuse_weights<<<256, 256, 0, stream>>>(Wk_tag,   bk_tag,   rel_att + 1 * TSLICE, FW + 2 * 65536, FB + 2 * 256);
  fuse_weights<<<256, 256, 0, stream>>>(Wv_tag,   bv_tag,   rel_msg + 1 * TSLICE, FW + 3 * 65536, FB + 3 * 256);
  fuse_weights<<<256, 256, 0, stream>>>(Wk_video, bk_video, rel_att + 2 * TSLICE, FW + 4 * 65536, FB + 4 * 256);
  fuse_weights<<<256, 256, 0, stream>>>(Wv_video, bv_video, rel_msg + 2 * TSLICE, FW + 5 * 65536, FB + 5 * 256);

  // 2) node GEMMs (WMMA f32), v_* land directly in d_out
  auto gemm = [&](const float* X, const float* W, const float* b, float* C, int M) {
    dim3 g((unsigned)((M + 63) / 64), 4);
    gemm_wmma_f32<<<g, 128, 0, stream>>>(X, W, b, C, M);
  };
  gemm(h_video,  FW + 0 * 65536, FB + 0 * 256, k_fb,        Nv);
  gemm(h_video,  FW + 1 * 65536, FB + 1 * 256, out + O_vfb, Nv);
  gemm(h0_video, Wq_video,       bq_video,     q_v,         Nv);
  gemm(h_tag,    FW + 2 * 65536, FB + 2 * 256, k_st,        Nt);
  gemm(h_tag,    FW + 3 * 65536, FB + 3 * 256, out + O_vst, Nt);
  gemm(h0_tag,   Wq_tag,         bq_tag,       q_t,         Nt);
  gemm(h_video,  FW + 4 * 65536, FB + 4 * 256, k_ht,        Nv);
  gemm(h_video,  FW + 5 * 65536, FB + 5 * 256, out + O_vht, Nv);

  // 3) zero softmax tables (m_v,s_v,m_t,s_t are contiguous: (Nv+Nt)*16 words)
  {
    int n = (Nv + Nt) * 16;
    zero_u32<<<(n + 255) / 256, 256, 0, stream>>>(m_v, n);
  }

  const float scale = 0.1767766952966369f;  // 1/sqrt(32)
  // 4) scores + segment max (video dsts: fb only; tag dsts: st+ht jointly)
  edge_scores<<<(E_fb * 32 + 255) / 256, 256, 0, stream>>>(q_v, k_fb, fb_src, fb_dst, rel_pri + 0,  scale, E_fb, out + O_tfb, m_v);
  edge_scores<<<(E_st * 32 + 255) / 256, 256, 0, stream>>>(q_t, k_st, st_src, st_dst, rel_pri + 8,  scale, E_st, out + O_tst, m_t);
  edge_scores<<<(E_ht * 32 + 255) / 256, 256, 0, stream>>>(q_t, k_ht, ht_src, ht_dst, rel_pri + 16, scale, E_ht, out + O_tht, m_t);
  // 5) exp + segment sum
  edge_exp<<<(E_fb * 8 + 255) / 256, 256, 0, stream>>>(fb_dst, m_v, out + O_tfb, s_v, E_fb);
  edge_exp<<<(E_st * 8 + 255) / 256, 256, 0, stream>>>(st_dst, m_t, out + O_tst, s_t, E_st);
  edge_exp<<<(E_ht * 8 + 255) / 256, 256, 0, stream>>>(ht_dst, m_t, out + O_tht, s_t, E_ht);
  // 6) normalize
  edge_norm<<<(E_fb * 8 + 255) / 256, 256, 0, stream>>>(fb_dst, s_v, out + O_tfb, E_fb);
  edge_norm<<<(E_st * 8 + 255) / 256, 256, 0, stream>>>(st_dst, s_t, out + O_tst, E_st);
  edge_norm<<<(E_ht * 8 + 255) / 256, 256, 0, stream>>>(ht_dst, s_t, out + O_tht, E_ht);
}